// SANeuronTorch_24352464569938
// MI455X (gfx1250) — compile-verified
//
#include <hip/hip_runtime.h>
#include <stdint.h>

// SA-neuron forward scan for MI455X (gfx1250).
//
// Shapes: input [B=16, T=2000, F=1024] f32.
// Output: concat( imem_trace [B, T+1, F] f32 , spikes [B, T+1, F] as f32 0/1 ).
//
// One thread per (b,f) neuron; strictly-serial time scan (threshold/reset
// nonlinearity forbids parallel prefix). Memory-bound: ~393 MB @ 23.3 TB/s
// ~ 17 us. Only 512 wave32s of parallelism exist, so latency hiding comes
// from a per-wave 32-deep ring of GLOBAL_LOAD_ASYNC_TO_LDS_B32 prefetches
// (CDNA5 async path, ASYNCcnt), consumed with ds_load_b32 (AS3 pointer,
// DScnt only — NOT a flat load). No barriers: each wave owns its columns.

#define T_STEPS 2000
#define F_DIM   1024
#define B_DIM   16
#define BLOCK   64            // 2 wave32s per block; 256 blocks total
#define NSLOT   32            // prefetch depth (ASYNCcnt is 6-bit; 32 is max usable pow2)
#define MAIN_END (T_STEPS - NSLOT)

// ---- f32 constants, evaluated in double from the reference defaults ----
#define CURRENT_SCALE_F 1.0e-12f
#define ONE_MINUS_A_F   0.2f                     // 1-0.8 rounds to f32 0.2
#define DT_S_F          7.428571428571428e-06f   // TAU_S/20
#define INV_TAU_M_F     6.730769230769231e+03f   // 1/1.4857142857142857e-4
#define INV_TAU_AHP_F   2.1538461538461538e+03f  // 1/4.642857142857143e-4
#define INV_TAU_REF_F   2.1538461538461538e+05f  // 1/4.642857142857143e-6
#define Z_TH_F          2.5e-09f
#define I_TH_AHP_F      1.66e-11f
#define I_REF_INC_F     1.6e-09f                 // I_TAU_REF added on spike
#define R_F             332.0f

typedef __attribute__((address_space(3))) float       lds_f32;
typedef __attribute__((address_space(1))) const float gbl_f32;

// CDNA5 async global->LDS copy: per-lane 4B, tracked by ASYNCcnt.
__device__ __forceinline__ void async_ld_b32(lds_f32* lp, gbl_f32* gp) {
    asm volatile("global_load_async_to_lds_b32 %0, %1, off"
                 :: "v"(lp), "v"(gp)
                 : "memory");
}
// True LDS read (ds_load_b32): volatile AS(3) load, DScnt-tracked.
__device__ __forceinline__ float lds_read(lds_f32* lp) {
    return *(volatile lds_f32*)lp;
}
#define WAIT_ASYNC(n) asm volatile("s_wait_asynccnt %0" :: "n"(n) : "memory")
#define WAIT_DS0()    asm volatile("s_wait_dscnt 0" ::: "memory")

// One SA-neuron Euler step. Returns the imem_trace value (z_next * R),
// updates state in place, writes spike (0/1) to `spike`.
__device__ __forceinline__ float sa_step(float x, float& z, float& iahp,
                                         float& iref, float& spike) {
    const float i_in  = x * CURRENT_SCALE_F;
    const float i_net = i_in - iahp - iref;
    const float dz    = (i_net - ONE_MINUS_A_F * z) * INV_TAU_M_F;
    float zn          = z + DT_S_F * dz;
    const bool fired  = (zn >= Z_TH_F);
    zn                = fired ? 0.0f : zn;
    float ia = iahp + DT_S_F * (-(iahp * INV_TAU_AHP_F));
    float ir = iref + DT_S_F * (-(iref * INV_TAU_REF_F));
    ia = fired ? (ia + I_TH_AHP_F)  : ia;
    ir = fired ? (ir + I_REF_INC_F) : ir;
    z = zn; iahp = ia; iref = ir;
    spike = fired ? 1.0f : 0.0f;
    return zn * R_F;
}

__global__ __launch_bounds__(BLOCK)
void sa_scan_kernel(const float* __restrict__ in, float* __restrict__ out) {
    const int tid = threadIdx.x;
    const int f   = blockIdx.x * BLOCK + tid;
    const int b   = blockIdx.y;

    __shared__ float buf[NSLOT][BLOCK];   // 8 KB ring; wave-private columns

    const float* gin = in + ((size_t)b * T_STEPS) * F_DIM + f;
    float* imem = out + ((size_t)b * (T_STEPS + 1)) * F_DIM + f;
    float* spk  = out + ((size_t)B_DIM * (T_STEPS + 1)) * F_DIM
                      + ((size_t)b * (T_STEPS + 1)) * F_DIM + f;

    __builtin_nontemporal_store(0.0f, imem);   // imem_trace[:,0,:] = 0
    imem += F_DIM;                             // now at row t+1 = 1

    // ---- prologue: fill the async ring (NSLOT outstanding per wave) ----
    {
        const float* g = gin;
#pragma unroll
        for (int p = 0; p < NSLOT; ++p) {
            async_ld_b32((lds_f32*)&buf[p][tid], (gbl_f32*)g);
            g += F_DIM;
        }
    }
    const float* gnext = gin + (size_t)NSLOT * F_DIM;

    float z = 0.0f, iahp = 0.0f, iref = 0.0f, spike_val = 0.0f;

    int t = 0;
    // ---- main loop: wait oldest slot, consume via ds_load, refill ----
#pragma unroll 4
    for (; t < MAIN_END; ++t) {
        WAIT_ASYNC(NSLOT - 1);                 // async loads complete in order
        const int slot = t & (NSLOT - 1);
        lds_f32* lp = (lds_f32*)&buf[slot][tid];
        const float x = lds_read(lp);          // ds_load_b32
        WAIT_DS0();                            // drain ds read before refill
        async_ld_b32(lp, (gbl_f32*)gnext);
        gnext += F_DIM;

        const float im = sa_step(x, z, iahp, iref, spike_val);
        __builtin_nontemporal_store(im, imem);
        __builtin_nontemporal_store(spike_val, spk);
        imem += F_DIM; spk += F_DIM;
    }

    // ---- tail: all loads already issued; drain once, then no waits ----
    WAIT_ASYNC(0);
#pragma unroll 4
    for (; t < T_STEPS; ++t) {
        const int slot = t & (NSLOT - 1);
        const float x = lds_read((lds_f32*)&buf[slot][tid]);
        const float im = sa_step(x, z, iahp, iref, spike_val);
        __builtin_nontemporal_store(im, imem);
        __builtin_nontemporal_store(spike_val, spk);
        imem += F_DIM; spk += F_DIM;
    }

    // spikes[:, T, :] = fired_{T-1}
    __builtin_nontemporal_store(spike_val, spk);
}

extern "C" void kernel_launch(void* const* d_in, const int* in_sizes, int n_in,
                              void* d_out, int out_size, void* d_ws, size_t ws_size,
                              hipStream_t stream) {
    (void)in_sizes; (void)n_in; (void)out_size; (void)d_ws; (void)ws_size;
    const float* in = (const float*)d_in[0];
    float* out      = (float*)d_out;
    dim3 grid(F_DIM / BLOCK, B_DIM);   // (16, 16) = 256 blocks
    dim3 block(BLOCK);                 // 64 threads = 2 wave32s
    sa_scan_kernel<<<grid, block, 0, stream>>>(in, out);
}